// GCN_9131100472028
// MI455X (gfx1250) — compile-verified
//
#include <hip/hip_runtime.h>
#include <hip/hip_bf16.h>

typedef __attribute__((ext_vector_type(2))) float v2f;
typedef __attribute__((ext_vector_type(8))) float v8f;

// ---------------- degree / normalization ----------------

__global__ void k_deg_init(float* __restrict__ deg, int n) {
    int i = blockIdx.x * blockDim.x + threadIdx.x;
    if (i < n) deg[i] = 1.0f;                      // self-loop contribution
}

__global__ void k_deg_edges(const int* __restrict__ col, float* __restrict__ deg, int e) {
    int i = blockIdx.x * blockDim.x + threadIdx.x;
    if (i < e) atomicAdd(&deg[col[i]], 1.0f);
}

__global__ void k_dinv(float* __restrict__ deg, int n) {
    int i = blockIdx.x * blockDim.x + threadIdx.x;
    if (i < n) deg[i] = rsqrtf(deg[i]);            // deg >= 1 always
}

// ---------------- GEMM1: h = x @ W1   (N x 64) = (N x 64)(64 x 64) ----------------
// One wave per 16x16 output tile, f32 WMMA, K accumulated in 16 steps of 4.

__global__ void k_gemm1(const float* __restrict__ x, const float* __restrict__ W1,
                        float* __restrict__ h, int n) {
    const int wave  = blockIdx.x * (blockDim.x >> 5) + (threadIdx.x >> 5);
    const int ntile_total = (n >> 4) * 4;          // 4 tiles along the 64-wide N dim
    if (wave >= ntile_total) return;               // wave-uniform: EXEC stays all-1s
    const int m0 = (wave >> 2) << 4;
    const int n0 = (wave & 3) << 4;
    const int lane = threadIdx.x & 31;
    const int l16  = lane & 15;                    // A: row M ; B: col N
    const int half = lane >> 4;

    v8f c = {};
    const float* arow = x  + (size_t)(m0 + l16) * 64;
    const float* bcol = W1 + (n0 + l16);
    #pragma unroll
    for (int k0 = 0; k0 < 64; k0 += 4) {
        const int ka = k0 + 2 * half;
        v2f a = *(const v2f*)(arow + ka);          // A[m][ka], A[m][ka+1]
        v2f b;
        b.x = bcol[(size_t)ka * 64];               // B[ka][n]
        b.y = bcol[(size_t)(ka + 1) * 64];         // B[ka+1][n]
        c = __builtin_amdgcn_wmma_f32_16x16x4_f32(false, a, false, b,
                                                  (short)0, c, false, false);
    }
    float* drow = h + n0 + l16;
    #pragma unroll
    for (int r = 0; r < 8; ++r)
        drow[(size_t)(m0 + r + 8 * half) * 64] = c[r];
}

// ---------------- layer-1 aggregation ----------------
// self-loop init: agg1[v] = h[v] * dinv[v]^2   (32 lanes x float2 per node)

__global__ void k_self1(const float* __restrict__ h, const float* __restrict__ dinv,
                        float* __restrict__ agg1, int n) {
    long long t = (long long)blockIdx.x * blockDim.x + threadIdx.x;
    int node = (int)(t >> 5);
    if (node >= n) return;
    int f2 = ((int)t & 31) << 1;
    float s = dinv[node]; s *= s;
    v2f v = *(const v2f*)(h + (size_t)node * 64 + f2);
    v2f r = {v.x * s, v.y * s};
    *(v2f*)(agg1 + (size_t)node * 64 + f2) = r;
}

// edge scatter: agg1[c] += h[r] * dinv[r]*dinv[c]   (32 lanes x float2 per edge)

__global__ void k_edge1(const int* __restrict__ row, const int* __restrict__ col,
                        const float* __restrict__ dinv, const float* __restrict__ h,
                        float* __restrict__ agg1, int e) {
    long long t = (long long)blockIdx.x * blockDim.x + threadIdx.x;
    int ei = (int)(t >> 5);
    if (ei >= e) return;
    int f2 = ((int)t & 31) << 1;
    int r = row[ei], c = col[ei];
    float nm = dinv[r] * dinv[c];
    v2f v = *(const v2f*)(h + (size_t)r * 64 + f2);
    atomicAdd(agg1 + (size_t)c * 64 + f2,     v.x * nm);
    atomicAdd(agg1 + (size_t)c * 64 + f2 + 1, v.y * nm);
}

__global__ void k_relubias(float* __restrict__ agg1, const float* __restrict__ b1, int n) {
    int i = blockIdx.x * blockDim.x + threadIdx.x;
    if (i >= n * 64) return;
    float v = agg1[i] + b1[i & 63];
    agg1[i] = v > 0.0f ? v : 0.0f;
}

// ---------------- GEMM2: h2 = h1 @ W2pad   (N x 8 stored, cols 6..15 zero) ----------------

__global__ void k_gemm2(const float* __restrict__ h1, const float* __restrict__ W2,
                        float* __restrict__ h2, int n) {
    __shared__ float w2s[64 * 16];                 // W2 zero-padded to 16 cols
    for (int i = threadIdx.x; i < 64 * 16; i += blockDim.x) {
        int k = i >> 4, nn = i & 15;
        w2s[i] = (nn < 6) ? W2[k * 6 + nn] : 0.0f;
    }
    __syncthreads();

    const int wave = blockIdx.x * (blockDim.x >> 5) + (threadIdx.x >> 5);
    if (wave >= (n >> 4)) return;                  // wave-uniform exit
    const int m0 = wave << 4;
    const int lane = threadIdx.x & 31;
    const int l16  = lane & 15;
    const int half = lane >> 4;

    v8f c = {};
    const float* arow = h1 + (size_t)(m0 + l16) * 64;
    #pragma unroll
    for (int k0 = 0; k0 < 64; k0 += 4) {
        const int ka = k0 + 2 * half;
        v2f a = *(const v2f*)(arow + ka);
        v2f b;
        b.x = w2s[ka * 16 + l16];
        b.y = w2s[(ka + 1) * 16 + l16];
        c = __builtin_amdgcn_wmma_f32_16x16x4_f32(false, a, false, b,
                                                  (short)0, c, false, false);
    }
    if (l16 < 8) {                                 // store padded stride-8 rows
        float* drow = h2 + l16;
        #pragma unroll
        for (int r = 0; r < 8; ++r)
            drow[(size_t)(m0 + r + 8 * half) * 8] = c[r];
    }
}

// ---------------- layer-2 aggregation ----------------

__global__ void k_self2(const float* __restrict__ h2, const float* __restrict__ dinv,
                        float* __restrict__ agg2, int n) {
    int i = blockIdx.x * blockDim.x + threadIdx.x;
    if (i >= n * 8) return;
    float s = dinv[i >> 3]; s *= s;
    agg2[i] = h2[i] * s;
}

__global__ void k_edge2(const int* __restrict__ row, const int* __restrict__ col,
                        const float* __restrict__ dinv, const float* __restrict__ h2,
                        float* __restrict__ agg2, int e) {
    long long t = (long long)blockIdx.x * blockDim.x + threadIdx.x;
    int ei = (int)(t >> 3);                        // 8 lanes per edge
    if (ei >= e) return;
    int f = (int)t & 7;
    int r = row[ei], c = col[ei];
    float nm = dinv[r] * dinv[c];
    float v = h2[(size_t)r * 8 + f] * nm;
    if (f < 6) atomicAdd(agg2 + (size_t)c * 8 + f, v);
}

// ---------------- final mean reduction ----------------

__global__ void k_outinit(float* __restrict__ out, const float* __restrict__ b2) {
    int i = threadIdx.x;
    if (i < 6) out[i] = b2[i];                     // mean(agg2 + b2) = mean(agg2) + b2
}

__global__ void k_reduce(const float* __restrict__ agg2, float* __restrict__ out,
                         int n, float invn) {
    __shared__ float red[6];
    if (threadIdx.x < 6) red[threadIdx.x] = 0.0f;
    __syncthreads();
    float s0 = 0, s1 = 0, s2 = 0, s3 = 0, s4 = 0, s5 = 0;
    for (int i = blockIdx.x * blockDim.x + threadIdx.x; i < n;
         i += gridDim.x * blockDim.x) {
        const float* p = agg2 + (size_t)i * 8;
        s0 += p[0]; s1 += p[1]; s2 += p[2];
        s3 += p[3]; s4 += p[4]; s5 += p[5];
    }
    atomicAdd(&red[0], s0); atomicAdd(&red[1], s1); atomicAdd(&red[2], s2);
    atomicAdd(&red[3], s3); atomicAdd(&red[4], s4); atomicAdd(&red[5], s5);
    __syncthreads();
    if (threadIdx.x < 6) atomicAdd(&out[threadIdx.x], red[threadIdx.x] * invn);
}

// ---------------- launch ----------------

extern "C" void kernel_launch(void* const* d_in, const int* in_sizes, int n_in,
                              void* d_out, int out_size, void* d_ws, size_t ws_size,
                              hipStream_t stream) {
    const float* x  = (const float*)d_in[0];
    const int*   ei = (const int*)d_in[1];
    const float* W1 = (const float*)d_in[2];
    const float* b1 = (const float*)d_in[3];
    const float* W2 = (const float*)d_in[4];
    const float* b2 = (const float*)d_in[5];
    float* out = (float*)d_out;

    const int n = in_sizes[0] / 64;      // 100,000 (multiple of 16)
    const int e = in_sizes[1] / 2;       // 3,200,000
    const int* row = ei;
    const int* col = ei + e;

    // workspace layout (floats):
    //   [0, n)        dinv (deg in-place)
    //   [n, 65n)      h  (layer-1 GEMM out) -> dead after relu; reused for h2/agg2
    //   [65n, 129n)   agg1 / h1
    float* ws   = (float*)d_ws;
    float* dinv = ws;
    float* h    = ws + (size_t)n;
    float* agg1 = h + (size_t)n * 64;
    float* h2   = h;                     // N x 8 (padded), reuses dead h
    float* agg2 = h + (size_t)n * 8;     // N x 8

    const int B = 256;
    auto blks = [](long long work, int blk) { return (unsigned)((work + blk - 1) / blk); };

    k_deg_init <<<blks(n, B), B, 0, stream>>>(dinv, n);
    k_deg_edges<<<blks(e, B), B, 0, stream>>>(col, dinv, e);
    k_dinv     <<<blks(n, B), B, 0, stream>>>(dinv, n);

    const int tiles1 = (n >> 4) * 4;     // waves for GEMM1
    k_gemm1<<<blks(tiles1, 8), B, 0, stream>>>(x, W1, h, n);

    k_self1<<<blks((long long)n * 32, B), B, 0, stream>>>(h, dinv, agg1, n);
    k_edge1<<<blks((long long)e * 32, B), B, 0, stream>>>(row, col, dinv, h, agg1, e);
    k_relubias<<<blks((long long)n * 64, B), B, 0, stream>>>(agg1, b1, n);

    const int tiles2 = n >> 4;
    k_gemm2<<<blks(tiles2, 8), B, 0, stream>>>(agg1, W2, h2, n);

    k_self2<<<blks((long long)n * 8, B), B, 0, stream>>>(h2, dinv, agg2, n);
    k_edge2<<<blks((long long)e * 8, B), B, 0, stream>>>(row, col, dinv, h2, agg2, e);

    k_outinit<<<1, 32, 0, stream>>>(out, b2);
    k_reduce <<<256, B, 0, stream>>>(agg2, out, n, 1.0f / (float)n);
}